// BinsChamferLoss_multi_16200616640819
// MI455X (gfx1250) — compile-verified
//
#include <hip/hip_runtime.h>
#include <hip/hip_bf16.h>

typedef float v2f __attribute__((ext_vector_type(2)));
typedef float v8f __attribute__((ext_vector_type(8)));

#define NB      2
#define NBINS   257
#define P       256      // bin centers per (n,l)
#define LPATCH  16       // 4x4 patches
#define KP      112      // patch edge
#define Q       (KP*KP)  // 12544 samples per patch
#define QTILES  (Q/16)   // 784
#define HW      448
#define THREADS 256
#define WAVES   8

__device__ __forceinline__ unsigned umin32(unsigned a, unsigned b) {
    return a < b ? a : b;   // lowers to v_min_u32 / v_min3_u32
}

__device__ __forceinline__ float blockReduceSum(float v, float* sRed) {
    const int tid = threadIdx.x;
    sRed[tid] = v;
    __syncthreads();
    for (int s = THREADS / 2; s > 0; s >>= 1) {
        if (tid < s) sRed[tid] += sRed[tid + s];
        __syncthreads();
    }
    float r = sRed[0];
    __syncthreads();
    return r;
}

__global__ void __launch_bounds__(THREADS)
chamfer_block_kernel(const float* __restrict__ bins,
                     const float* __restrict__ tgt,
                     float* __restrict__ ws) {
    __shared__ float xs[Q];          // 50 KB patch samples (sentinel for invalid)
    __shared__ float a0s[P];         // c^2
    __shared__ float a1s[P];         // -2c
    __shared__ float minY[P];        // per-center min distance (unique-writer)
    __shared__ float sRed[THREADS];
    extern __shared__ unsigned colMinU[];   // Q entries (dynamic, 50 KB): per-sample min bits

    const int tid  = threadIdx.x;
    const int lane = tid & 31;
    const int wave = tid >> 5;
    const int col  = lane & 15;
    const bool hi  = lane >= 16;
    const int blk  = blockIdx.x;     // 0..31 == (n, l)
    const int n    = blk >> 4;
    const int l    = blk & 15;

    // ---- Stage A: centers -> WMMA A-row coefficients [c^2, -2c, 1, 0] ----
    {
        float b0 = bins[(n * NBINS + tid    ) * LPATCH + l];
        float b1 = bins[(n * NBINS + tid + 1) * LPATCH + l];
        float c  = 0.5f * (b0 + b1);
        a0s[tid] = c * c;
        a1s[tid] = -2.0f * c;
    }
    for (int q = tid; q < Q; q += THREADS) colMinU[q] = 0x7f800000u;  // +inf bits

    // ---- Stage B: unfold patch l of image n into LDS ----
    {
        const int br = (l >> 2) * KP;
        const int bc = (l & 3) * KP;
        const float* img = tgt + (size_t)n * HW * HW;
        for (int q = tid; q < Q; q += THREADS) {
            int i = q / KP, j = q - i * KP;
            float x = img[(br + i) * HW + (bc + j)];
            // invalid -> large-magnitude sentinel: distance ~1e36, never wins a min
            xs[q] = (x > 0.0f) ? x : -1.0e18f;
        }
    }
    __syncthreads();

    // ---- Hoist A operands: this wave owns p-tiles {2*wave, 2*wave+1} ----
    // A layout (16x4 f32): lanes 0-15 hold K=0,1 (c^2, -2c); lanes 16-31 hold K=2,3 (1, 0)
    v2f a0v, a1v;
    {
        const int p0 = (2 * wave)     * 16 + col;
        const int p1 = (2 * wave + 1) * 16 + col;
        float c00 = a0s[p0], c01 = a1s[p0];   // broadcast reads (same addr per half)
        float c10 = a0s[p1], c11 = a1s[p1];
        a0v.x = hi ? 1.0f : c00;  a0v.y = hi ? 0.0f : c01;
        a1v.x = hi ? 1.0f : c10;  a1v.y = hi ? 0.0f : c11;
    }

    // per-row running min over q, kept as IEEE bits: uint order == float order for
    // non-negatives; negative round-off bits (sign set) auto-lose every min.
    unsigned rowU0[8], rowU1[8];
    #pragma unroll
    for (int r = 0; r < 8; ++r) { rowU0[r] = 0x7f800000u; rowU1[r] = 0x7f800000u; }

    const v8f czero = {};

    // every wave sweeps all q-tiles; 2 WMMAs per tile, EXEC uniform throughout
    #pragma unroll 2
    for (int qt = 0; qt < QTILES; ++qt) {
        const float x = xs[qt * 16 + col];
        // B column basis [1, x, x^2, 0]: lanes 0-15 hold K=0,1; lanes 16-31 hold K=2,3
        v2f b;
        b.x = hi ? x * x : 1.0f;
        b.y = hi ? 0.0f  : x;

        v8f d0 = __builtin_amdgcn_wmma_f32_16x16x4_f32(
            false, a0v, false, b, (short)0, czero, false, false);
        v8f d1 = __builtin_amdgcn_wmma_f32_16x16x4_f32(
            false, a1v, false, b, (short)0, czero, false, false);

        unsigned cm = 0x7f800000u;   // min over this lane's 16 rows for column q
        #pragma unroll
        for (int r = 0; r < 8; ++r) {
            unsigned u0 = __float_as_uint(d0[r]);
            unsigned u1 = __float_as_uint(d1[r]);
            rowU0[r] = umin32(rowU0[r], u0);
            rowU1[r] = umin32(rowU1[r], u1);
            cm = umin32(cm, umin32(u0, u1));
        }
        // fold this wave's 32-row partial into the per-sample min (deterministic)
        atomicMin(&colMinU[qt * 16 + col], cm);
    }

    // ---- one-time cross-lane reduction of the register row accumulators ----
    #pragma unroll
    for (int r = 0; r < 8; ++r) {
        unsigned m0 = rowU0[r], m1 = rowU1[r];
        m0 = umin32(m0, (unsigned)__shfl_xor((int)m0, 1, 32));
        m0 = umin32(m0, (unsigned)__shfl_xor((int)m0, 2, 32));
        m0 = umin32(m0, (unsigned)__shfl_xor((int)m0, 4, 32));
        m0 = umin32(m0, (unsigned)__shfl_xor((int)m0, 8, 32));
        m1 = umin32(m1, (unsigned)__shfl_xor((int)m1, 1, 32));
        m1 = umin32(m1, (unsigned)__shfl_xor((int)m1, 2, 32));
        m1 = umin32(m1, (unsigned)__shfl_xor((int)m1, 4, 32));
        m1 = umin32(m1, (unsigned)__shfl_xor((int)m1, 8, 32));
        if (col == 0) {   // rows are uniquely owned by this wave -> plain stores
            const int off = (hi ? 8 : 0) + r;
            minY[(2 * wave)     * 16 + off] = fmaxf(__uint_as_float(m0), 0.0f);
            minY[(2 * wave + 1) * 16 + off] = fmaxf(__uint_as_float(m1), 0.0f);
        }
    }
    __syncthreads();

    // ---- cham_y: sum per-sample nearest-center distance over valid samples ----
    float sumY = 0.0f, cntY = 0.0f;
    for (int q = tid; q < Q; q += THREADS) {
        if (xs[q] > 0.0f) {
            sumY += fmaxf(__uint_as_float(colMinU[q]), 0.0f);
            cntY += 1.0f;
        }
    }

    float sumX = blockReduceSum(minY[tid], sRed);  // sum_p min_q d
    float totY = blockReduceSum(sumY, sRed);       // sum_{valid q} min_p d
    float totC = blockReduceSum(cntY, sRed);       // valid count

    if (tid == 0) {
        float chamX = sumX * (1.0f / 256.0f);
        float chamY = totY / fmaxf(totC, 1.0f);
        ws[blk] = (totC > 0.0f) ? (chamX + chamY) : 0.0f;
    }
}

__global__ void chamfer_finalize(const float* __restrict__ ws,
                                 float* __restrict__ out) {
    float v = ws[threadIdx.x];  // 32 per-patch values, one wave
    v += __shfl_xor(v, 1, 32);
    v += __shfl_xor(v, 2, 32);
    v += __shfl_xor(v, 4, 32);
    v += __shfl_xor(v, 8, 32);
    v += __shfl_xor(v, 16, 32);
    if (threadIdx.x == 0) out[0] = v * (1.0f / 32.0f);
}

extern "C" void kernel_launch(void* const* d_in, const int* in_sizes, int n_in,
                              void* d_out, int out_size, void* d_ws, size_t ws_size,
                              hipStream_t stream) {
    const float* bins = (const float*)d_in[0];   // (2, 257, 4, 4) fp32
    const float* tgt  = (const float*)d_in[1];   // (2, 1, 448, 448) fp32
    float* out = (float*)d_out;                  // scalar fp32
    float* ws  = (float*)d_ws;                   // 32 per-patch partials

    const size_t dynLds = (size_t)Q * sizeof(unsigned);   // 50 KB column-min array
    chamfer_block_kernel<<<dim3(NB * LPATCH), dim3(THREADS), dynLds, stream>>>(bins, tgt, ws);
    chamfer_finalize<<<dim3(1), dim3(32), 0, stream>>>(ws, out);
}